// RevScaledDotProductAttention_65223373357266
// MI455X (gfx1250) — compile-verified
//
#include <hip/hip_runtime.h>
#include <hip/hip_bf16.h>

typedef __attribute__((ext_vector_type(2))) float v2f;
typedef __attribute__((ext_vector_type(8))) float v8f;

#define B 64
#define N 4096
#define D 128
#define NCHUNK 16
#define CHUNK (N / NCHUNK)              // 256 rows of k/v per block
#define INV_T 0.08838834764831845f      // 1/sqrt(128)

// ---------------------------------------------------------------------------
// Phase 1: partial S[b] += (K/T)^T @ V  for a chunk of n-rows.
// Block = 256 threads (8 waves). Wave w owns d_k tile-row w (16 rows of S),
// loops over the 8 d_v tile-cols. WMMA f32 16x16x4 (full precision).
// A-frag layout (32-bit A 16x4): lane L holds A[M=L%16][K=2*(L/16)+r] in vgpr r.
// C/D layout: vgpr r, lane L -> row = r + 8*(L/16), col = L%16.
// ---------------------------------------------------------------------------
__global__ void __launch_bounds__(256)
ktv_partial(const float* __restrict__ K, const float* __restrict__ V,
            float* __restrict__ S) {
  const int chunk = blockIdx.x;
  const int batch = blockIdx.y;
  const int wave  = threadIdx.x >> 5;
  const int lane  = threadIdx.x & 31;
  const int half  = lane >> 4;          // 0: lanes 0-15, 1: lanes 16-31
  const int lm    = lane & 15;

  const float* Kb = K + (size_t)batch * N * D;
  const float* Vb = V + (size_t)batch * N * D;
  const int n0 = chunk * CHUNK;
  const int i  = wave;                  // d_k tile row (0..7)

  v8f acc[8] = {};

  for (int kb = 0; kb < CHUNK; kb += 4) {
    const int na = n0 + kb + 2 * half;  // first of this lane's two K-slices
    // prefetch ahead in the K stream
    __builtin_prefetch(&Kb[(size_t)(na + 8) * D + 16 * i + lm], 0, 1);

    v2f a;
    a.x = Kb[(size_t)(na + 0) * D + 16 * i + lm] * INV_T;
    a.y = Kb[(size_t)(na + 1) * D + 16 * i + lm] * INV_T;

#pragma unroll
    for (int j = 0; j < 8; ++j) {       // d_v tile col
      v2f bf;
      bf.x = Vb[(size_t)(na + 0) * D + 16 * j + lm];
      bf.y = Vb[(size_t)(na + 1) * D + 16 * j + lm];
      acc[j] = __builtin_amdgcn_wmma_f32_16x16x4_f32(
          false, a, false, bf, (short)0, acc[j], false, false);
    }
  }

  float* Sb = S + (size_t)batch * D * D;
#pragma unroll
  for (int j = 0; j < 8; ++j) {
#pragma unroll
    for (int r = 0; r < 8; ++r) {
      const int row = 16 * i + r + 8 * half;
      const int col = 16 * j + lm;
      atomicAdd(&Sb[row * D + col], acc[j][r]);
    }
  }
}

// ---------------------------------------------------------------------------
// Phase 2: row softmax over the 128-wide d_v axis. One wave32 per row,
// float4 per lane, shfl_xor reductions. Writes attn into d_out tuple slot 2.
// ---------------------------------------------------------------------------
__global__ void __launch_bounds__(256)
softmax_rows(const float* __restrict__ S, float* __restrict__ attn) {
  const int wave = threadIdx.x >> 5;
  const int lane = threadIdx.x & 31;
  const int row  = blockIdx.x * 8 + wave;   // 0 .. B*D-1

  const float4* src = (const float4*)(S + (size_t)row * D);
  float4 v = src[lane];

  float m = fmaxf(fmaxf(v.x, v.y), fmaxf(v.z, v.w));
#pragma unroll
  for (int off = 16; off > 0; off >>= 1)
    m = fmaxf(m, __shfl_xor(m, off, 32));

  v.x = __expf(v.x - m);
  v.y = __expf(v.y - m);
  v.z = __expf(v.z - m);
  v.w = __expf(v.w - m);

  float s = v.x + v.y + v.z + v.w;
#pragma unroll
  for (int off = 16; off > 0; off >>= 1)
    s += __shfl_xor(s, off, 32);

  const float inv = 1.0f / s;
  v.x *= inv; v.y *= inv; v.z *= inv; v.w *= inv;

  ((float4*)(attn + (size_t)row * D))[lane] = v;
}

// ---------------------------------------------------------------------------
// Phase 3: O[b] = Q[b] @ attn[b]. Block handles 128 Q-rows; attn (64 KB) is
// staged into LDS with gfx1250 async DMA loads (GLOBAL_LOAD_ASYNC_TO_LDS_B128,
// tracked by ASYNCcnt) instead of a VGPR round-trip, then a 128x128 @ 128x128
// WMMA tile GEMM. Wave w owns output row-tile w.
// ---------------------------------------------------------------------------
__global__ void __launch_bounds__(256)
q_attn(const float* __restrict__ Q, const float* __restrict__ attn,
       float* __restrict__ O) {
  __shared__ float sA[D * D];           // 64 KB (WGP LDS is 320 KB)

  const int batch = blockIdx.y;
  const int n0    = blockIdx.x * 128;
  const int wave  = threadIdx.x >> 5;
  const int lane  = threadIdx.x & 31;
  const int half  = lane >> 4;
  const int lm    = lane & 15;

  // ---- async DMA: 64 KB of attn[b] -> LDS (16 B per lane per op) ----
  {
    const unsigned long long gbase =
        (unsigned long long)(uintptr_t)(attn + (size_t)batch * D * D);
    const unsigned ldsbase = (unsigned)(uintptr_t)&sA[0];
    const unsigned toff    = (unsigned)threadIdx.x * 16u;
#pragma unroll
    for (int it = 0; it < 16; ++it) {   // 16 iters * 256 thr * 16 B = 64 KB
      const unsigned off = toff + (unsigned)it * 4096u;
      const unsigned long long ga = gbase + off;
      const unsigned la = ldsbase + off;
      asm volatile("global_load_async_to_lds_b128 %0, %1, off"
                   :: "v"(la), "v"(ga) : "memory");
    }
    asm volatile("s_wait_asynccnt 0" ::: "memory");
  }
  __syncthreads();

  const float* Qb = Q + ((size_t)batch * N + n0) * D;
  const int i = wave;

  v8f acc[8] = {};

  for (int kb = 0; kb < D; kb += 4) {
    const int kk = kb + 2 * half;
    // A[m][k] = Q[n0+16i+m][k]; the two K values are contiguous -> float2 load
    v2f a = *(const v2f*)&Qb[(size_t)(16 * i + lm) * D + kk];

#pragma unroll
    for (int j = 0; j < 8; ++j) {
      v2f bf;
      bf.x = sA[(kk + 0) * D + 16 * j + lm];
      bf.y = sA[(kk + 1) * D + 16 * j + lm];
      acc[j] = __builtin_amdgcn_wmma_f32_16x16x4_f32(
          false, a, false, bf, (short)0, acc[j], false, false);
    }
  }

  float* Ob = O + ((size_t)batch * N + n0) * D;
#pragma unroll
  for (int j = 0; j < 8; ++j) {
#pragma unroll
    for (int r = 0; r < 8; ++r) {
      const int row = 16 * i + r + 8 * half;
      const int col = 16 * j + lm;
      Ob[(size_t)row * D + col] = acc[j][r];
    }
  }
}

// ---------------------------------------------------------------------------
extern "C" void kernel_launch(void* const* d_in, const int* in_sizes, int n_in,
                              void* d_out, int out_size, void* d_ws, size_t ws_size,
                              hipStream_t stream) {
  const float* q = (const float*)d_in[0];
  const float* k = (const float*)d_in[1];
  const float* v = (const float*)d_in[2];

  float* out      = (float*)d_out;                       // [B, N, D]
  float* attn_out = out + (size_t)B * N * D;             // [B, D, D] (tuple #2)
  float* S        = (float*)d_ws;                        // f32 accumulator, 4 MB

  hipMemsetAsync(d_ws, 0, (size_t)B * D * D * sizeof(float), stream);

  ktv_partial<<<dim3(NCHUNK, B), 256, 0, stream>>>(k, v, S);
  softmax_rows<<<(B * D) / 8, 256, 0, stream>>>(S, attn_out);
  q_attn<<<dim3(N / 128, B), 256, 0, stream>>>(q, attn_out, out);
}